// NetworkClass_SelfOrg_31018253812099
// MI455X (gfx1250) — compile-verified
//
#include <hip/hip_runtime.h>

#define N_IN   2048
#define NN     512
#define BATCH  64
#define KSPLIT 8
#define KCHUNK (N_IN / KSPLIT)   // 256

__device__ __constant__ const float kALPHA = 1.0f - 0.05f / 10.0f;  // 0.995
__device__ __constant__ const float kBETA  = 1.0f - 0.05f / 2.0f;   // 0.975
__device__ __constant__ const float kVTH   = 2.0f;

typedef __attribute__((ext_vector_type(2))) float v2f;
typedef __attribute__((ext_vector_type(4))) float v4f;
typedef __attribute__((ext_vector_type(8))) float v8f;

// ---------------------------------------------------------------------------
// Kernel 1: z_out_new = BETA*z_out + z  (to ws + output segment 2), and
// materialize the leave-one-out recurrent weight matrix
//   Wfull[m][n] = (m==n) ? 0 : w[N_IN + m - (m>n)][n]
// ---------------------------------------------------------------------------
__global__ void prep_kernel(const float* __restrict__ z,
                            const float* __restrict__ z_out,
                            const float* __restrict__ w,
                            float* __restrict__ zon,
                            float* __restrict__ wfull,
                            float* __restrict__ out_zon) {
    int i = blockIdx.x * blockDim.x + threadIdx.x;    // 0 .. NN*NN-1
    int m = i >> 9;
    int n = i & (NN - 1);
    float wv = 0.0f;
    if (m != n) {
        int row = N_IN + m - ((m > n) ? 1 : 0);
        wv = w[row * NN + n];
    }
    wfull[i] = wv;
    if (i < BATCH * NN) {
        float zo = kBETA * z_out[i] + z[i];
        zon[i]     = zo;
        out_zon[i] = zo;
    }
}

// ---------------------------------------------------------------------------
// Kernel 2: HBM-streaming reduction  drive1[b,n] = sum_k x[b,k,n]*w[k,n]
// k split into KSPLIT chunks; partials stored deterministically in ws.
// Contiguous axis is n -> b128 loads are fully coalesced.
// x (256 MiB, read-once, > 192 MB L2) is loaded non-temporally so it does
// not evict w (4 MiB, reused by all 64 batches) from L2.
// ---------------------------------------------------------------------------
__global__ void drive_kernel(const float* __restrict__ x,
                             const float* __restrict__ w,
                             float* __restrict__ part) {
    int blk = blockIdx.x;
    int c = blk >> 6;          // k-chunk 0..KSPLIT-1
    int b = blk & 63;          // batch 0..63
    int n4 = threadIdx.x * 4;  // 128 threads cover 512 n

    const v4f* xp = (const v4f*)(x + ((size_t)(b * N_IN + c * KCHUNK)) * NN + n4);
    const v4f* wp = (const v4f*)(w + (size_t)(c * KCHUNK) * NN + n4);

    v4f acc = {0.f, 0.f, 0.f, 0.f};
#pragma unroll 4
    for (int k = 0; k < KCHUNK; ++k) {
        v4f xv = __builtin_nontemporal_load(xp + (size_t)k * (NN / 4));  // stream, NT
        v4f wv = wp[(size_t)k * (NN / 4)];                               // L2-resident, RT
        acc.x = fmaf(xv.x, wv.x, acc.x);
        acc.y = fmaf(xv.y, wv.y, acc.y);
        acc.z = fmaf(xv.z, wv.z, acc.z);
        acc.w = fmaf(xv.w, wv.w, acc.w);
    }
    *(v4f*)(part + (size_t)blk * NN + n4) = acc;
}

// ---------------------------------------------------------------------------
// Kernel 3: drive_rec = z_out_new(64x512) @ Wfull(512x512) via
// V_WMMA_F32_16X16X4_F32 (one wave per 16x16 tile), then finalize:
//   v_new = ALPHA*v + drive1 + drive_rec - V_TH*z ; z_new = (v_new > V_TH)
// A layout (16x4): lane = M(0..15) twice; VGPR0/1 hold K = koff, koff+1 with
// koff = 0 (lanes 0-15) or 2 (lanes 16-31). B layout (4x16) symmetric.
// C/D: VGPR i = row (i | i+8), col = lane&15.
// ---------------------------------------------------------------------------
__global__ void wmma_finalize_kernel(const float* __restrict__ zon,
                                     const float* __restrict__ wfull,
                                     const float* __restrict__ part,
                                     const float* __restrict__ v,
                                     const float* __restrict__ z,
                                     float* __restrict__ out) {
    int wave = (blockIdx.x * blockDim.x + threadIdx.x) >> 5;  // 0..127
    int lane = threadIdx.x & 31;
    int bt = wave >> 5;   // 0..3   batch tile (rows of 16)
    int nt = wave & 31;   // 0..31  neuron tile (cols of 16)

    int mrow = lane & 15;
    int koff = (lane >> 4) << 1;  // 0 or 2

    const float* arow = zon + (size_t)(bt * 16 + mrow) * NN;          // A[m][k]
    const float* bcol = wfull + nt * 16 + (lane & 15);                // B[k][n], n fixed

    v8f acc = {};
    for (int kb = 0; kb < NN; kb += 4) {
        v2f a;
        a.x = arow[kb + koff];
        a.y = arow[kb + koff + 1];
        v2f bb;
        bb.x = bcol[(size_t)(kb + koff) * NN];
        bb.y = bcol[(size_t)(kb + koff + 1) * NN];
        acc = __builtin_amdgcn_wmma_f32_16x16x4_f32(
            /*neg_a=*/false, a, /*neg_b=*/false, bb,
            /*c_mod=*/(short)0, acc, /*reuse_a=*/false, /*reuse_b=*/false);
    }

    int n = nt * 16 + (lane & 15);
#pragma unroll
    for (int i = 0; i < 8; ++i) {
        int b = bt * 16 + ((lane < 16) ? i : (i + 8));
        float d1 = part[(size_t)b * NN + n];          // chunk 0
#pragma unroll
        for (int c = 1; c < KSPLIT; ++c)
            d1 += part[(size_t)(c * BATCH + b) * NN + n];
        float vn = kALPHA * v[b * NN + n] + d1 + acc[i] - kVTH * z[b * NN + n];
        out[b * NN + n] = vn;
        out[BATCH * NN + b * NN + n] = ((vn - kVTH) > 0.0f) ? 1.0f : 0.0f;
    }
}

extern "C" void kernel_launch(void* const* d_in, const int* in_sizes, int n_in,
                              void* d_out, int out_size, void* d_ws, size_t ws_size,
                              hipStream_t stream) {
    const float* x     = (const float*)d_in[0];
    const float* v     = (const float*)d_in[1];
    const float* z     = (const float*)d_in[2];
    const float* z_out = (const float*)d_in[3];
    const float* w     = (const float*)d_in[4];
    float* out = (float*)d_out;

    float* ws    = (float*)d_ws;
    float* zon   = ws;                  // BATCH*NN          = 32768 floats
    float* wfull = zon + BATCH * NN;    // NN*NN             = 262144 floats
    float* part  = wfull + NN * NN;     // KSPLIT*BATCH*NN   = 262144 floats

    // 1) elementwise update + materialize LOO weight matrix
    prep_kernel<<<(NN * NN) / 256, 256, 0, stream>>>(z, z_out, w, zon, wfull,
                                                     out + 2 * BATCH * NN);
    // 2) bandwidth-bound streaming reduction over x (256 MiB, the roofline term)
    drive_kernel<<<KSPLIT * BATCH, NN / 4, 0, stream>>>(x, w, part);
    // 3) WMMA recurrent GEMM + partial reduction + finalize
    wmma_finalize_kernel<<<16, 256, 0, stream>>>(zon, wfull, part, v, z, out);
}